// ContextKnowledgeEncoder_45320494908014
// MI455X (gfx1250) — compile-verified
//
#include <hip/hip_runtime.h>
#include <hip/hip_bf16.h>
#include <math.h>

// Problem constants from the reference
#define N_    32
#define K_    32
#define TK_   128
#define TS_   512
#define D_    512
#define TFULL_ (TK_ + TS_)   // 640

typedef __attribute__((ext_vector_type(2))) float v2f;
typedef __attribute__((ext_vector_type(8))) float v8f;

// ---------------------------------------------------------------------------
// Kernel 1: USE embeddings (masked sum over T of gathered E rows, scaled).
// One block per (n, k) pair; block index n*(K+1)+k, k==K_ means "context".
// 256 threads, each covers d and d+256 (coalesced 1KB bursts per E row).
// ---------------------------------------------------------------------------
__global__ void use_sum_kernel(const int* __restrict__ src_tokens,   // [N,TS]
                               const int* __restrict__ know_tokens,  // [N,K,TK]
                               const float* __restrict__ E,          // [V,D]
                               float* __restrict__ ctx_use,          // [N,D]
                               float* __restrict__ know_use)         // [N,K,D]
{
    __shared__ int toks[TS_];
    const int bid = blockIdx.x;
    const int n   = bid / (K_ + 1);
    const int k   = bid % (K_ + 1);
    const int tid = threadIdx.x;
    const bool is_ctx = (k == K_);
    const int  T = is_ctx ? TS_ : TK_;
    const int* tokp = is_ctx ? (src_tokens + (size_t)n * TS_)
                             : (know_tokens + ((size_t)n * K_ + k) * TK_);
    for (int t = tid; t < T; t += blockDim.x) toks[t] = tokp[t];
    __syncthreads();

    float acc0 = 0.f, acc1 = 0.f;
    int len = 0;
    const int d = tid;
    for (int t = 0; t < T; ++t) {
        const int tok = toks[t];
        if (tok != 0) {
            ++len;
            const float* row = E + (size_t)tok * D_;
            acc0 += row[d];
            acc1 += row[d + 256];
        }
    }
    const float l = (float)(len < 1 ? 1 : len);
    const float scale = rsqrtf(l) * rsqrtf((float)D_);
    float* out = is_ctx ? (ctx_use + (size_t)n * D_)
                        : (know_use + ((size_t)n * K_ + k) * D_);
    out[d]       = acc0 * scale;
    out[d + 256] = acc1 * scale;
}

// ---------------------------------------------------------------------------
// Kernel 2: ck_attn[n,k] = dot(know_use[n,k,:], ctx_use[n,:]) via
// V_WMMA_F32_16X16X4_F32. One block (2 waves) per example n; wave w owns
// k-rows [16w, 16w+16). B replicates the context d-chunk across all 16
// columns; column 0 of D is the true result (lane 0: M=0..7, lane 16: M=8..15).
// ---------------------------------------------------------------------------
__global__ void ck_attn_wmma_kernel(const float* __restrict__ know_use, // [N,K,D]
                                    const float* __restrict__ ctx_use,  // [N,D]
                                    float* __restrict__ ck_attn)        // [N,K]
{
    const int n    = blockIdx.x;
    const int wave = threadIdx.x >> 5;   // k-tile: rows 16*wave .. 16*wave+15
    const int lane = threadIdx.x & 31;
    const int half = lane >> 4;          // 0 -> K=0,1 ; 1 -> K=2,3
    const int l16  = lane & 15;

    // A: lane L<16 holds row M=L, K=0/1; lane L>=16 holds row M=L-16, K=2/3
    const float* Arow = know_use + ((size_t)n * K_ + (size_t)(wave * 16 + l16)) * D_;
    const float* Bvec = ctx_use + (size_t)n * D_;

    v8f c = {};
    #pragma unroll 4
    for (int d0 = 0; d0 < D_; d0 += 4) {
        v2f a, b;
        a.x = Arow[d0 + 2 * half + 0];
        a.y = Arow[d0 + 2 * half + 1];
        b.x = Bvec[d0 + 2 * half + 0];   // B[k][j] identical for all columns j
        b.y = Bvec[d0 + 2 * half + 1];
        c = __builtin_amdgcn_wmma_f32_16x16x4_f32(
                /*neg_a=*/false, a, /*neg_b=*/false, b,
                /*c_mod=*/(short)0, c, /*reuse_a=*/false, /*reuse_b=*/false);
    }

    // Column 0 of the 16x16 result: lane 0 has M=0..7 in c[0..7], lane 16 has M=8..15
    if (l16 == 0) {
        float* out = ck_attn + (size_t)n * K_ + wave * 16 + half * 8;
        #pragma unroll
        for (int r = 0; r < 8; ++r) out[r] = c[r];
    }
}

// ---------------------------------------------------------------------------
// Kernel 3: per-example argmax over K (first-index tie break, matching
// jnp.argmax), honoring use_cs_ids dynamically. One wave32 per example.
// ---------------------------------------------------------------------------
__global__ void argmax_kernel(const float* __restrict__ ck_attn, // [N,K]
                              const int* __restrict__ cs_ids,    // [N]
                              const int* __restrict__ use_cs_ids,// [1]
                              int* __restrict__ cs_eff)          // [N]
{
    const int n    = blockIdx.x;
    const int lane = threadIdx.x;  // K_ == 32 == warpSize
    float v = ck_attn[(size_t)n * K_ + lane];
    int  idx = lane;
    #pragma unroll
    for (int off = 16; off > 0; off >>= 1) {
        const float ov = __shfl_xor(v, off, 32);
        const int   oi = __shfl_xor(idx, off, 32);
        if (ov > v || (ov == v && oi < idx)) { v = ov; idx = oi; }
    }
    if (lane == 0)
        cs_eff[n] = (use_cs_ids[0] != 0) ? cs_ids[n] : idx;
}

// ---------------------------------------------------------------------------
// Kernel 4: assemble full_enc = concat(cs_encoded, context_encoded) and
// full_mask. One block per (n, t) output row; 256 threads cover D=512.
// Re-gathers only the *chosen* knowledge sentence (never materializes
// know_encoded [N,K,TK,D]).
// ---------------------------------------------------------------------------
__global__ void assemble_kernel(const int* __restrict__ src_tokens,
                                const int* __restrict__ know_tokens,
                                const float* __restrict__ E,
                                const int* __restrict__ cs_eff,
                                float* __restrict__ full_enc,   // [N,TFULL,D]
                                float* __restrict__ full_mask)  // [N,TFULL]
{
    const int bid = blockIdx.x;          // n * TFULL_ + t
    const int n   = bid / TFULL_;
    const int t   = bid % TFULL_;
    const int d   = threadIdx.x;         // 0..255

    int tok;
    if (t < TK_) {
        const int kk = cs_eff[n];
        tok = know_tokens[((size_t)n * K_ + kk) * TK_ + t];
    } else {
        tok = src_tokens[(size_t)n * TS_ + (t - TK_)];
    }
    const bool m = (tok != 0);
    float e0 = 0.f, e1 = 0.f;
    if (m) {
        const float* row = E + (size_t)tok * D_;
        e0 = row[d];
        e1 = row[d + 256];
    }
    float* out = full_enc + (size_t)bid * D_;
    out[d]       = e0;
    out[d + 256] = e1;
    if (d == 0) full_mask[bid] = m ? 1.0f : 0.0f;
}

// ---------------------------------------------------------------------------
extern "C" void kernel_launch(void* const* d_in, const int* in_sizes, int n_in,
                              void* d_out, int out_size, void* d_ws, size_t ws_size,
                              hipStream_t stream)
{
    const int*   src_tokens  = (const int*)d_in[0];   // [N,TS]
    const int*   know_tokens = (const int*)d_in[1];   // [N,K,TK]
    /* d_in[2] = ck_mask (unused by reference) */
    const int*   cs_ids      = (const int*)d_in[3];   // [N]
    const int*   use_cs_ids  = (const int*)d_in[4];   // [1]
    const float* E           = (const float*)d_in[5]; // [V,D]

    float* out       = (float*)d_out;
    float* full_enc  = out;                                    // N*TFULL*D
    float* full_mask = out + (size_t)N_ * TFULL_ * D_;         // N*TFULL
    float* ck_attn   = full_mask + (size_t)N_ * TFULL_;        // N*K

    float* ctx_use  = (float*)d_ws;                            // N*D
    float* know_use = ctx_use + (size_t)N_ * D_;               // N*K*D
    int*   cs_eff   = (int*)(know_use + (size_t)N_ * K_ * D_); // N

    use_sum_kernel<<<N_ * (K_ + 1), 256, 0, stream>>>(
        src_tokens, know_tokens, E, ctx_use, know_use);

    ck_attn_wmma_kernel<<<N_, 64, 0, stream>>>(know_use, ctx_use, ck_attn);

    argmax_kernel<<<N_, 32, 0, stream>>>(ck_attn, cs_ids, use_cs_ids, cs_eff);

    assemble_kernel<<<N_ * TFULL_, 256, 0, stream>>>(
        src_tokens, know_tokens, E, cs_eff, full_enc, full_mask);
}